// ConditionalLogisticRegression_76261439308149
// MI455X (gfx1250) — compile-verified
//
#include <hip/hip_runtime.h>
#include <hip/hip_bf16.h>

// Conditional logistic regression: logits = X @ W^T + b, then segmented
// (per-stratum) numerically-stable softmax over 65536 sorted segments.
//
// Memory-bound problem (~1.1 GiB traffic, ~50us at 23.3 TB/s). The 4M x 64
// mat-vec runs on the f32 WMMA path (V_WMMA_F32_16X16X4_F32) so the logits
// keep full f32 precision; segment max/sum state (128K words) lives in L2
// via atomics.

#define DIM 64
#define NUM_SEG 65536

typedef float v2f __attribute__((ext_vector_type(2)));
typedef float v8f __attribute__((ext_vector_type(8)));

// Monotone f32 <-> u32 order-preserving map so unsigned atomicMax implements
// IEEE float max (works for negatives; 0u acts as -inf sentinel).
__device__ __forceinline__ unsigned f32_ordered(float f) {
  unsigned u = __float_as_uint(f);
  return (u & 0x80000000u) ? ~u : (u | 0x80000000u);
}
__device__ __forceinline__ float ordered_f32(unsigned u) {
  return (u & 0x80000000u) ? __uint_as_float(u & 0x7FFFFFFFu)
                           : __uint_as_float(~u);
}

// ---------------------------------------------------------------------------
// Kernel A: per-wave 16-row tile; logits via V_WMMA_F32_16X16X4_F32.
//   A (16x4 f32): lane L<16 holds X[row L][k0+0..1], lane L>=16 holds K=2..3.
//   B (4x16 f32): W chunk in column N=0 only (lane 0 carries K=0..1 pair,
//                 lane 16 carries K=2..3 pair; all other lanes zero).
//   D (16x16 f32): column N=0 -> lane 0 has rows 0..7 in c[0..7],
//                  lane 16 has rows 8..15.
// ---------------------------------------------------------------------------
__global__ void clr_logits_wmma(const float* __restrict__ X,
                                const int* __restrict__ seg,
                                const float* __restrict__ W,
                                const float* __restrict__ bias,
                                float* __restrict__ logits,
                                unsigned* __restrict__ segmax,
                                long long n_rows) {
  const int lane = threadIdx.x & 31;
  const long long waveId =
      (long long)blockIdx.x * (blockDim.x >> 5) + (threadIdx.x >> 5);
  const long long rowBase = waveId * 16;
  if (rowBase + 16 > n_rows) return;  // wave-uniform: EXEC stays all-1s

  const int r  = lane & 15;        // row within tile (A) / column N (B,D)
  const int kh = (lane >> 4) << 1; // K sub-offset: 0 (lanes 0-15) or 2 (16-31)

  const float* xrow = X + (rowBase + r) * DIM + kh;

  v8f c = {};
#pragma unroll
  for (int k0 = 0; k0 < DIM; k0 += 4) {
    const float2 a2 = *(const float2*)(xrow + k0);
    const float2 w2 = *(const float2*)(W + k0 + kh);  // L0-resident, broadcast
    v2f a;  a.x = a2.x;  a.y = a2.y;
    v2f bm; bm.x = (r == 0) ? w2.x : 0.0f;  // only column N=0 is real
            bm.y = (r == 0) ? w2.y : 0.0f;
    c = __builtin_amdgcn_wmma_f32_16x16x4_f32(
        /*neg_a=*/false, a, /*neg_b=*/false, bm,
        /*c_mod=*/(short)0, c, /*reuse_a=*/false, /*reuse_b=*/false);
  }

  if (r == 0) {  // lanes 0 and 16 hold column N=0 of D
    const float bv = bias[0];
    const long long base = rowBase + ((long long)(lane >> 4) << 3);
#pragma unroll
    for (int i = 0; i < 8; ++i) {
      const float lg = c[i] + bv;
      logits[base + i] = lg;
      atomicMax(&segmax[seg[base + i]], f32_ordered(lg));
    }
  }
}

// ---------------------------------------------------------------------------
// Kernel B: ex = exp(logit - segmax[seg]) in-place; denom[seg] += ex.
// ---------------------------------------------------------------------------
__global__ void clr_exp_sum(const int* __restrict__ seg,
                            const unsigned* __restrict__ segmax,
                            float* __restrict__ out,   // in: logits, out: ex
                            float* __restrict__ denom,
                            long long n) {
  const long long i = (long long)blockIdx.x * blockDim.x + threadIdx.x;
  if (i >= n) return;
  const int s = seg[i];
  const float m  = ordered_f32(segmax[s]);  // finite for any non-empty segment
  const float ex = expf(out[i] - m);
  out[i] = ex;
  atomicAdd(&denom[s], ex);
}

// ---------------------------------------------------------------------------
// Kernel C: out = ex / denom[seg].
// ---------------------------------------------------------------------------
__global__ void clr_norm(const int* __restrict__ seg,
                         const float* __restrict__ denom,
                         float* __restrict__ out,
                         long long n) {
  const long long i = (long long)blockIdx.x * blockDim.x + threadIdx.x;
  if (i >= n) return;
  out[i] = out[i] / denom[seg[i]];
}

extern "C" void kernel_launch(void* const* d_in, const int* in_sizes, int n_in,
                              void* d_out, int out_size, void* d_ws, size_t ws_size,
                              hipStream_t stream) {
  const float* X   = (const float*)d_in[0];
  const int*   seg = (const int*)d_in[1];   // harness: integer -> const int*
  const float* W   = (const float*)d_in[2];
  const float* b   = (const float*)d_in[3];
  float* out = (float*)d_out;

  unsigned* segmax = (unsigned*)d_ws;
  float*    denom  = (float*)((char*)d_ws + (size_t)NUM_SEG * sizeof(unsigned));

  const long long n = (long long)in_sizes[1];  // N rows

  // segmax=0 encodes -inf under the ordered map; denom starts at 0.
  hipMemsetAsync(d_ws, 0, (size_t)NUM_SEG * (sizeof(unsigned) + sizeof(float)),
                 stream);

  // A: 256 threads = 8 waves/block, 16 rows/wave -> 128 rows/block.
  const int blocksA = (int)((n + 127) / 128);
  clr_logits_wmma<<<blocksA, 256, 0, stream>>>(X, seg, W, b, out, segmax, n);

  const int blocksE = (int)((n + 255) / 256);
  clr_exp_sum<<<blocksE, 256, 0, stream>>>(seg, segmax, out, denom, n);
  clr_norm<<<blocksE, 256, 0, stream>>>(seg, denom, out, n);
}